// Linear_85358180041313
// MI455X (gfx1250) — compile-verified
//
#include <hip/hip_runtime.h>
#include <hip/hip_bf16.h>

// ---------------------------------------------------------------------------
// Social-force sim step, N=4096.
// Hot loop: 16.7M pair interactions. d^2 tiles computed with
// V_WMMA_F32_16X16X4_F32 via a bordered Gram trick; per-pair transcendentals
// (v_rsq_f32 + v_exp_f32) in VALU; rf reduced as p_i*S - T.
// Grid = 2048 single-wave workgroups so tile indices are SGPRs and the
// diagonal tile peels into uniform scalar control flow (no EXEC churn).
// ---------------------------------------------------------------------------

#define N        4096
#define NT       256          // N/16 tiles
#define SLICES   8
#define TPS      32           // tiles per slice (NT/SLICES)
#define DT       0.4f
#define PED_SPEED 1.0f
#define K_ATTR   2.0f
#define PED_MASS 60.0f
#define A_COST   5.0f
#define B_COST   2.0f
#define EPS      1e-8f
#define LOG2E    1.44269504088896340736f
// w*dist = ALPHA*exp((0.6-dist)/0.71)  ->  exp2(C0 + C1*dist)
#define C0       4.5411070083146506f   // log2(10) + 0.6*log2e/0.71
#define C1      (-2.0319648463224837f) // -log2e/0.71

#define OUT_COST  (4*N)
#define OUT_STACK (4*N + 1)

typedef __attribute__((ext_vector_type(2))) float v2f;
typedef __attribute__((ext_vector_type(8))) float v8f;

// ---------------------------------------------------------------------------
// Prep: build the per-lane B operand array for the Gram WMMA.
// B layout (4x16 f32): lanes 0-15 hold rows K=0,1 (VGPR0,VGPR1),
// lanes 16-31 hold rows K=2,3.  Per J-tile t, lane l:
//   l<16 : (px_j, py_j)        (K0,K1)
//   l>=16: (1.0,  rj^2)        (K2,K3)
// ---------------------------------------------------------------------------
__global__ void prep_kernel(const float* __restrict__ state,
                            float2* __restrict__ bop) {
  int gid = blockIdx.x * 256 + threadIdx.x;       // NT*32 = 8192 threads
  if (gid >= NT * 32) return;
  int lane = gid & 31;
  int t    = gid >> 5;
  int j    = t * 16 + (lane & 15);
  float px = state[j * 4 + 0];
  float py = state[j * 4 + 1];
  float2 v;
  if (lane < 16) { v.x = px;   v.y = py; }
  else           { v.x = 1.0f; v.y = fmaf(px, px, py * py); }
  bop[gid] = v;
}

// ---------------------------------------------------------------------------
// Per-tile elementwise: d2 -> w, accumulate S/Tx/Ty in C-layout.
// DIAG=true only for the one tile containing i==j (exact (1-eye) masking).
// ---------------------------------------------------------------------------
template <bool DIAG>
__device__ __forceinline__ void accum_tile(v8f d, float2 pj, int jdiag,
                                           float (&S)[8], float (&Tx)[8],
                                           float (&Ty)[8]) {
#pragma unroll
  for (int r = 0; r < 8; ++r) {
    float d2   = d[r];                               // = |pi-pj|^2 + eps
    float rsq  = __builtin_amdgcn_rsqf(d2);          // 1/dist
    float dist = d2 * rsq;                           // dist
    float w    = __builtin_amdgcn_exp2f(fmaf(dist, C1, C0)) * rsq;
    if (DIAG) w = (r == jdiag) ? 0.0f : w;           // mask self-interaction
    S[r]  += w;
    Tx[r]  = fmaf(w, pj.x, Tx[r]);
    Ty[r]  = fmaf(w, pj.y, Ty[r]);
  }
}

// ---------------------------------------------------------------------------
// Pairwise kernel: one wave per block. block = (I-tile, J-slice):
//   itile = blockIdx.x >> 3  (0..255),  slice = blockIdx.x & 7.
// A operand (16x4 f32): lanes 0-15 K0/K1 = (-2px_i, -2py_i),
//                       lanes 16-31 K2/K3 = (ri^2+eps, 1).
// D[m,n] = -2 pi.pj + ri^2 + eps + rj^2 = |pi-pj|^2 + eps.
// ---------------------------------------------------------------------------
__global__ void __launch_bounds__(32)
pair_forces_kernel(const float* __restrict__ state,
                   const float2* __restrict__ bop,
                   float* __restrict__ pS,
                   float* __restrict__ pTx,
                   float* __restrict__ pTy) {
  const int lane  = threadIdx.x;        // 0..31
  const int itile = blockIdx.x >> 3;    // SGPR
  const int slice = blockIdx.x & 7;     // SGPR
  const int n16   = lane & 15;
  const int hi    = lane >> 4;

  const int   i   = itile * 16 + n16;
  const float pix = state[i * 4 + 0];
  const float piy = state[i * 4 + 1];
  const float ri2 = fmaf(pix, pix, piy * piy);

  v2f a;
  a.x = hi ? (ri2 + EPS) : (-2.0f * pix);
  a.y = hi ? 1.0f        : (-2.0f * piy);

  float S[8]  = {0.f, 0.f, 0.f, 0.f, 0.f, 0.f, 0.f, 0.f};
  float Tx[8] = {0.f, 0.f, 0.f, 0.f, 0.f, 0.f, 0.f, 0.f};
  float Ty[8] = {0.f, 0.f, 0.f, 0.f, 0.f, 0.f, 0.f, 0.f};
  const v8f cz = {0.f, 0.f, 0.f, 0.f, 0.f, 0.f, 0.f, 0.f};

  auto fast_range = [&](int tb, int te) {
#pragma unroll 2
    for (int t = tb; t < te; ++t) {
      float2 bv = bop[t * 32 + lane];   // B operand (per-lane prebuilt)
      float2 pj = bop[t * 32 + n16];    // (px_j, py_j) for this lane's column
      v2f b; b.x = bv.x; b.y = bv.y;
      v8f d = __builtin_amdgcn_wmma_f32_16x16x4_f32(
          false, a, false, b, (short)0, cz, false, false);
      accum_tile<false>(d, pj, 0, S, Tx, Ty);
    }
  };

  const int t0 = slice * TPS, t1 = t0 + TPS;
  const int td = itile;                       // diagonal tile index (SGPR)
  const bool owns = (td >= t0) && (td < t1);  // uniform scalar condition

  fast_range(t0, owns ? td : t1);
  if (owns) {
    {  // diagonal tile, exact self-interaction masking
      float2 bv = bop[td * 32 + lane];
      float2 pj = bop[td * 32 + n16];
      v2f b; b.x = bv.x; b.y = bv.y;
      v8f d = __builtin_amdgcn_wmma_f32_16x16x4_f32(
          false, a, false, b, (short)0, cz, false, false);
      accum_tile<true>(d, pj, n16 - (hi << 3), S, Tx, Ty);
    }
    fast_range(td + 1, t1);
  }

  // Reduce over the j (lane) dimension within each 16-lane half.
#pragma unroll
  for (int r = 0; r < 8; ++r) {
#pragma unroll
    for (int k = 8; k >= 1; k >>= 1) {
      S[r]  += __shfl_xor(S[r],  k, 32);
      Tx[r] += __shfl_xor(Tx[r], k, 32);
      Ty[r] += __shfl_xor(Ty[r], k, 32);
    }
  }
  if (n16 == 0) {
#pragma unroll
    for (int r = 0; r < 8; ++r) {
      int ii = itile * 16 + (hi << 3) + r;   // C-layout: M = r + 8*hi
      pS [slice * N + ii] = S[r];
      pTx[slice * N + ii] = Tx[r];
      pTy[slice * N + ii] = Ty[r];
    }
  }
}

// ---------------------------------------------------------------------------
// Finalize per-agent: rf = (p*S - T)/m, attractive force, integrate, clamp,
// write output state + both halves of the stacked trajectory buffer.
// ---------------------------------------------------------------------------
__global__ void finalize_kernel(const float* __restrict__ state,
                                const float* __restrict__ goal,
                                const float* __restrict__ stacked_in,
                                const float* __restrict__ pS,
                                const float* __restrict__ pTx,
                                const float* __restrict__ pTy,
                                float* __restrict__ out) {
  int i = blockIdx.x * 256 + threadIdx.x;
  if (i >= N) return;
  float px = state[i * 4 + 0], py = state[i * 4 + 1];
  float vx = state[i * 4 + 2], vy = state[i * 4 + 3];

  float S = 0.f, Tx = 0.f, Ty = 0.f;
#pragma unroll
  for (int s = 0; s < SLICES; ++s) {
    S += pS[s * N + i]; Tx += pTx[s * N + i]; Ty += pTy[s * N + i];
  }
  const float invM = 1.0f / PED_MASS;
  float rfx = (px * S - Tx) * invM;
  float rfy = (py * S - Ty) * invM;

  float gx = goal[i * 2 + 0] - px, gy = goal[i * 2 + 1] - py;
  float invdg = __builtin_amdgcn_rsqf(fmaf(gx, gx, fmaf(gy, gy, EPS)));
  float afx = K_ATTR * (PED_SPEED * gx * invdg - vx);
  float afy = K_ATTR * (PED_SPEED * gy * invdg - vy);

  float Fx = rfx + afx, Fy = rfy + afy;
  float npx = px + vx * DT + 0.5f * Fx * (DT * DT);
  float npy = py + vy * DT + 0.5f * Fy * (DT * DT);
  float nvx = vx + Fx * DT, nvy = vy + Fy * DT;
  float invs = __builtin_amdgcn_rsqf(fmaf(nvx, nvx, fmaf(nvy, nvy, EPS)));
  float sc = fminf(1.0f, PED_SPEED * invs);
  nvx *= sc; nvy *= sc;

  out[i * 4 + 0] = npx; out[i * 4 + 1] = npy;
  out[i * 4 + 2] = nvx; out[i * 4 + 3] = nvy;

  // stacked = concat(stacked_in, state)
#pragma unroll
  for (int c = 0; c < 4; ++c)
    out[OUT_STACK + i * 4 + c] = stacked_in[i * 4 + c];
  out[OUT_STACK + 4 * N + i * 4 + 0] = px;
  out[OUT_STACK + 4 * N + i * 4 + 1] = py;
  out[OUT_STACK + 4 * N + i * 4 + 2] = vx;
  out[OUT_STACK + 4 * N + i * 4 + 3] = vy;
}

// ---------------------------------------------------------------------------
// Cost reduction: single 1024-thread block (32 waves), shuffle + LDS tree.
// Reads the new robot pose from out[0..1] (stream-ordered after finalize).
// ---------------------------------------------------------------------------
__global__ void cost_kernel(const float* __restrict__ state,
                            const float* __restrict__ cost_in,
                            const float* __restrict__ goal,
                            const float* __restrict__ observed,
                            float* __restrict__ out) {
  __shared__ float sdev[32], sbl[32];
  int tid = threadIdx.x;
  const float rx = out[0], ry = out[1];

  float dev = 0.f, bl = 0.f;
  for (int i = tid; i < N; i += 1024) {
    if (i == 0) continue;
    float ppx = out[i * 4 + 0], ppy = out[i * 4 + 1];
    float dx = ppx - observed[i * 4 + 0];
    float dy = ppy - observed[i * 4 + 1];
    dev += __builtin_amdgcn_sqrtf(fmaf(dx, dx, fmaf(dy, dy, EPS)));
    float ex = ppx - rx, ey = ppy - ry;
    float d = __builtin_amdgcn_sqrtf(fmaf(ex, ex, fmaf(ey, ey, EPS)));
    bl += __builtin_amdgcn_exp2f(-d * LOG2E);   // exp(-d/E_COST), E_COST=1
  }
#pragma unroll
  for (int k = 16; k >= 1; k >>= 1) {
    dev += __shfl_xor(dev, k, 32);
    bl  += __shfl_xor(bl,  k, 32);
  }
  if ((tid & 31) == 0) { sdev[tid >> 5] = dev; sbl[tid >> 5] = bl; }
  __syncthreads();
  if (tid < 32) {
    dev = sdev[tid]; bl = sbl[tid];
#pragma unroll
    for (int k = 16; k >= 1; k >>= 1) {
      dev += __shfl_xor(dev, k, 32);
      bl  += __shfl_xor(bl,  k, 32);
    }
    if (tid == 0) {
      float gx0 = goal[0], gy0 = goal[1];
      float d1x = state[0] - gx0, d1y = state[1] - gy0;
      float d2x = rx - gx0,       d2y = ry - gy0;
      float pg = __builtin_amdgcn_sqrtf(fmaf(d1x, d1x, fmaf(d1y, d1y, EPS))) -
                 __builtin_amdgcn_sqrtf(fmaf(d2x, d2x, fmaf(d2y, d2y, EPS)));
      out[OUT_COST] = cost_in[0] + A_COST * dev + B_COST * bl - pg;
    }
  }
}

// ---------------------------------------------------------------------------
extern "C" void kernel_launch(void* const* d_in, const int* in_sizes, int n_in,
                              void* d_out, int out_size, void* d_ws,
                              size_t ws_size, hipStream_t stream) {
  const float* state    = (const float*)d_in[0];  // (N,4)
  const float* cost     = (const float*)d_in[1];  // (1,)
  const float* stacked  = (const float*)d_in[2];  // (N,4)
  const float* goal     = (const float*)d_in[3];  // (N,2)
  const float* observed = (const float*)d_in[4];  // (N,4)
  float* out = (float*)d_out;
  float* ws  = (float*)d_ws;

  // ws layout (floats): bop[NT*32*2]=16384 | pS[8*N] | pTx[8*N] | pTy[8*N]
  float2* bop = (float2*)ws;
  float*  pS  = ws + 16384;
  float*  pTx = ws + 16384 + SLICES * N;
  float*  pTy = ws + 16384 + 2 * SLICES * N;

  prep_kernel       <<<32, 256, 0, stream>>>(state, bop);
  pair_forces_kernel<<<NT * SLICES, 32, 0, stream>>>(state, bop, pS, pTx, pTy);
  finalize_kernel   <<<N / 256, 256, 0, stream>>>(state, goal, stacked,
                                                  pS, pTx, pTy, out);
  cost_kernel       <<<1, 1024, 0, stream>>>(state, cost, goal, observed, out);
}